// GCL_Encoder_90340342104106
// MI455X (gfx1250) — compile-verified
//
#include <hip/hip_runtime.h>
#include <hip/hip_bf16.h>
#include <stdint.h>

#define NUM_SYM   50000
#define NUM_HERB  50000
#define N_NODES   (NUM_SYM + NUM_HERB)
#define DIM       128
#define N_EDGES   3200000
#define SCAN_TILE 1024
#define NUM_TILES ((N_NODES + SCAN_TILE - 1) / SCAN_TILE)   // 98
#define WPB       8                                          // waves per block

// ---------------------------------------------------------------------------
// Step 1: histogram of destination rows (int atomics, L2-resident counters).
// Edge stream is read once -> non-temporal so it doesn't pollute L2.
// ---------------------------------------------------------------------------
__global__ __launch_bounds__(256) void count_edges(const int* __restrict__ rows,
                                                   int* __restrict__ cnt) {
    int i = blockIdx.x * blockDim.x + threadIdx.x;
    if (i < N_EDGES) {
        int r = __builtin_nontemporal_load(&rows[i]);
        atomicAdd(&cnt[r], 1);
    }
}

// ---------------------------------------------------------------------------
// Step 2: exclusive scan of 100k counters (3 tiny kernels; cost is noise).
// ---------------------------------------------------------------------------
__global__ __launch_bounds__(256) void scan_tiles(const int* __restrict__ cnt,
                                                  int* __restrict__ tileSums) {
    __shared__ int s[256];
    const int t = threadIdx.x, b = blockIdx.x;
    const int base = b * SCAN_TILE + t * 4;
    int sum = 0;
#pragma unroll
    for (int k = 0; k < 4; ++k) {
        int idx = base + k;
        if (idx < N_NODES) sum += cnt[idx];
    }
    s[t] = sum;
    __syncthreads();
    if (t == 0) {
        int tot = 0;
        for (int i = 0; i < 256; ++i) tot += s[i];
        tileSums[b] = tot;
    }
}

__global__ void scan_tilesums(int* __restrict__ tileSums, int* __restrict__ row_ptr) {
    int run = 0;
    for (int i = 0; i < NUM_TILES; ++i) {
        int t = tileSums[i];
        tileSums[i] = run;
        run += t;
    }
    row_ptr[N_NODES] = N_EDGES;   // total is known by construction
}

// Writes both row_ptr (persistent) and the scatter cursors (consumed next).
__global__ __launch_bounds__(256) void scan_write(const int* __restrict__ cnt,
                                                  const int* __restrict__ tileSums,
                                                  int* __restrict__ row_ptr,
                                                  int* __restrict__ cursor) {
    __shared__ int s[256];
    __shared__ int p[256];
    const int t = threadIdx.x, b = blockIdx.x;
    const int base = b * SCAN_TILE + t * 4;
    int loc[4];
    int sum = 0;
#pragma unroll
    for (int k = 0; k < 4; ++k) {
        int idx = base + k;
        loc[k] = (idx < N_NODES) ? cnt[idx] : 0;
        sum += loc[k];
    }
    s[t] = sum;
    __syncthreads();
    if (t == 0) {
        int run = tileSums[b];
        for (int i = 0; i < 256; ++i) { p[i] = run; run += s[i]; }
    }
    __syncthreads();
    int pref = p[t];
#pragma unroll
    for (int k = 0; k < 4; ++k) {
        int idx = base + k;
        if (idx < N_NODES) {
            row_ptr[idx] = pref;
            cursor[idx]  = pref;
        }
        pref += loc[k];
    }
}

// ---------------------------------------------------------------------------
// Step 3: scatter edges into interleaved CSR slots: one 8B (col, val) pair
// per edge so the SpMM can fetch 32 edges with a single coalesced b64 load.
// ---------------------------------------------------------------------------
__global__ __launch_bounds__(256) void scatter_edges(const int* __restrict__ rows,
                                                     const int* __restrict__ cols,
                                                     const float* __restrict__ vals,
                                                     int* __restrict__ cursor,
                                                     int2* __restrict__ csr_pair) {
    int i = blockIdx.x * blockDim.x + threadIdx.x;
    if (i < N_EDGES) {
        int   r = __builtin_nontemporal_load(&rows[i]);
        int   c = __builtin_nontemporal_load(&cols[i]);
        float v = __builtin_nontemporal_load(&vals[i]);
        int pos = atomicAdd(&cursor[r], 1);
        int2 pr;
        pr.x = c;
        pr.y = __float_as_int(v);
        csr_pair[pos] = pr;
    }
}

// ---------------------------------------------------------------------------
// SpMM core: one wave32 per destination row. The row's edge list is streamed
// into a per-wave LDS staging buffer with CDNA5 async global->LDS DMA
// (global_load_async_to_lds_b64, ASYNCcnt), double-buffered: chunk i+1 is in
// flight while chunk i is consumed via uniform-address ds_load_b64 broadcasts.
// Per edge: 1 ds_load_b64 + 1 gathered global_load_b128 (L2 hit) + 4 FMAs.
// Lane l owns dims [4l, 4l+4); a single plain b128 store writes the row.
// ---------------------------------------------------------------------------
__device__ __forceinline__ void async_stage_b64(uint32_t lds_dst, uint64_t gaddr) {
    asm volatile("global_load_async_to_lds_b64 %0, %1, off"
                 :: "v"(lds_dst), "v"(gaddr) : "memory");
}

template <int LAYER>
__global__ __launch_bounds__(256) void spmm_rows(const int* __restrict__ row_ptr,
                                                 const int2* __restrict__ csr_pair,
                                                 const float* __restrict__ x0,   // sym  (L1) / e1 (L2)
                                                 const float* __restrict__ x1,   // herb (L1) / unused
                                                 const float* __restrict__ sym,
                                                 const float* __restrict__ herb,
                                                 const float* __restrict__ e1,
                                                 float* __restrict__ out) {
    __shared__ long long stage[WPB][2][32];
    const int w    = threadIdx.x >> 5;
    const int lane = threadIdx.x & 31;
    const int row  = blockIdx.x * WPB + w;
    if (row >= N_NODES) return;

    const int beg = row_ptr[row];
    const int end = row_ptr[row + 1];
    const int deg = end - beg;

    float ax = 0.f, ay = 0.f, az = 0.f, aw = 0.f;

    if (deg > 0) {
        const int nch = (deg + 31) >> 5;

        // Kick off chunk 0 (lanes past the end clamp to a valid slot).
        {
            int j = beg + lane;
            if (j >= end) j = end - 1;
            async_stage_b64((uint32_t)(uintptr_t)&stage[w][0][lane],
                            (uint64_t)(uintptr_t)(csr_pair + j));
        }

        for (int ch = 0; ch < nch; ++ch) {
            if (ch + 1 < nch) {
                int j = beg + ((ch + 1) << 5) + lane;
                if (j >= end) j = end - 1;
                async_stage_b64((uint32_t)(uintptr_t)&stage[w][(ch + 1) & 1][lane],
                                (uint64_t)(uintptr_t)(csr_pair + j));
                // Current chunk done when only the prefetched one is outstanding.
                asm volatile("s_wait_asynccnt 0x1" ::: "memory");
            } else {
                asm volatile("s_wait_asynccnt 0x0" ::: "memory");
            }

            int n = deg - (ch << 5);
            if (n > 32) n = 32;
            const long long* buf = &stage[w][ch & 1][0];
#pragma unroll 4
            for (int k = 0; k < n; ++k) {
                const long long pk = buf[k];                  // ds broadcast
                const int   c = (int)(uint32_t)pk;
                const float v = __int_as_float((int)(pk >> 32));
                const float* src;
                if (LAYER == 1)
                    src = (c < NUM_SYM) ? (x0 + (size_t)c * DIM)
                                        : (x1 + (size_t)(c - NUM_SYM) * DIM);
                else
                    src = x0 + (size_t)c * DIM;
                const float4 x = ((const float4*)src)[lane];
                ax = fmaf(v, x.x, ax);
                ay = fmaf(v, x.y, ay);
                az = fmaf(v, x.z, az);
                aw = fmaf(v, x.w, aw);
            }
        }
    }

    if (LAYER == 1) {
        float4 o;
        o.x = ax; o.y = ay; o.z = az; o.w = aw;
        ((float4*)(out + (size_t)row * DIM))[lane] = o;
    } else {
        // Fused (ego + e1 + e2) / 3 combine.
        const float* ego = (row < NUM_SYM) ? (sym + (size_t)row * DIM)
                                           : (herb + (size_t)(row - NUM_SYM) * DIM);
        const float4 e = ((const float4*)ego)[lane];
        const float4 m = ((const float4*)(e1 + (size_t)row * DIM))[lane];
        const float inv3 = 1.0f / 3.0f;
        float4 r;
        r.x = (e.x + m.x + ax) * inv3;
        r.y = (e.y + m.y + ay) * inv3;
        r.z = (e.z + m.z + az) * inv3;
        r.w = (e.w + m.w + aw) * inv3;
        ((float4*)(out + (size_t)row * DIM))[lane] = r;
    }
}

// ---------------------------------------------------------------------------
extern "C" void kernel_launch(void* const* d_in, const int* in_sizes, int n_in,
                              void* d_out, int out_size, void* d_ws, size_t ws_size,
                              hipStream_t stream) {
    const float* sym  = (const float*)d_in[0];
    const float* herb = (const float*)d_in[1];
    const int*   rows = (const int*)d_in[2];
    const int*   cols = (const int*)d_in[3];
    const float* vals = (const float*)d_in[4];
    float* out = (float*)d_out;

    // Workspace layout (256B-aligned chunks), ~78 MB total.
    char* p = (char*)d_ws;
    auto take = [&](size_t bytes) {
        char* q = p;
        p += (bytes + 255) & ~(size_t)255;
        return q;
    };
    float* e1         = (float*)take((size_t)N_NODES * DIM * sizeof(float)); // 51.2 MB
    int2*  csr_pair   = (int2*) take((size_t)N_EDGES * sizeof(int2));        // 25.6 MB
    int*   row_ptr    = (int*)  take((size_t)(N_NODES + 1) * sizeof(int));
    int*   cnt        = (int*)  take((size_t)N_NODES * sizeof(int));         // degree counts
    int*   cursor     = (int*)  take((size_t)N_NODES * sizeof(int));         // scatter cursors
    int*   tileSums   = (int*)  take((size_t)NUM_TILES * sizeof(int));

    const int EB = (N_EDGES + 255) / 256;       // edge-parallel blocks
    const int RB = (N_NODES + WPB - 1) / WPB;   // 1 wave per row

    hipMemsetAsync(cnt, 0, (size_t)N_NODES * sizeof(int), stream);
    count_edges   <<<EB,        256, 0, stream>>>(rows, cnt);
    scan_tiles    <<<NUM_TILES, 256, 0, stream>>>(cnt, tileSums);
    scan_tilesums <<<1,           1, 0, stream>>>(tileSums, row_ptr);
    scan_write    <<<NUM_TILES, 256, 0, stream>>>(cnt, tileSums, row_ptr, cursor);
    scatter_edges <<<EB,        256, 0, stream>>>(rows, cols, vals, cursor, csr_pair);

    // Layer 1: e1 = A @ ego            (ego = concat(sym, herb), virtual)
    spmm_rows<1><<<RB, 256, 0, stream>>>(row_ptr, csr_pair, sym, herb,
                                         sym, herb, e1, e1);
    // Layer 2: out = (ego + e1 + A@e1) / 3
    spmm_rows<2><<<RB, 256, 0, stream>>>(row_ptr, csr_pair, e1, nullptr,
                                         sym, herb, e1, out);
}